// RMILoss_8658654069145
// MI455X (gfx1250) — compile-verified
//
#include <hip/hip_runtime.h>
#include <math.h>

#define NUM_CLASSES 19
#define NN 4
#define HH 512
#define WW 512
#define PHH 129
#define PWW 129
#define NH 127
#define NW 127
#define KTOT (NH * NW)            // 16129 (logical K)
#define KPAD (NH * 128)           // 16256 (padded K domain: y = k>>7, x = k&127)
#define HALF_D 9
#define CLP_MIN 1e-06f
#define POS_ALPHA 0.0005
#define NPAIR (NN * NUM_CLASSES)  // 76
#define GWAVES 16                 // waves per Gram workgroup

typedef float v2f __attribute__((ext_vector_type(2)));
typedef float v8f __attribute__((ext_vector_type(8)));

// ---------------------------------------------------------------------------
// Kernel 0: zero global accumulators (ws is poisoned, atomics need clean init)
// ---------------------------------------------------------------------------
__global__ void rmi_init_kernel(double* bceSum, unsigned int* validCnt) {
    *bceSum = 0.0;
    *validCnt = 0u;
}

// ---------------------------------------------------------------------------
// Kernel 1: fused masked-BCE + sigmoid + one-hot + 4x4/stride-4/pad-2 avg pool.
// One thread per pooled cell per channel; pool windows are disjoint so every
// (n,c,h,w) logit is read exactly once -> BCE accumulated in the same pass.
// ---------------------------------------------------------------------------
__global__ void __launch_bounds__(256)
rmi_bce_pool_kernel(const float* __restrict__ logits,
                    const int* __restrict__ labels,
                    float* __restrict__ labPool,
                    float* __restrict__ prPool,
                    double* __restrict__ bceSum,
                    unsigned int* __restrict__ validCnt) {
    const int PP = PHH * PWW;
    const int TOT = NPAIR * PP;
    int idx = blockIdx.x * blockDim.x + threadIdx.x;

    float bceLocal = 0.f;
    unsigned int vLocal = 0u;

    if (idx < TOT) {
        int b    = idx / PP;
        int cell = idx - b * PP;
        int ph   = cell / PWW;
        int pw   = cell - ph * PWW;
        int n    = b / NUM_CLASSES;
        int c    = b - n * NUM_CLASSES;

        const float* lg = logits + (size_t)(n * NUM_CLASSES + c) * HH * WW;
        const int*   lb = labels + (size_t)n * HH * WW;

        float labSum = 0.f, prSum = 0.f;
        int h0 = 4 * ph - 2;
        int w0 = 4 * pw - 2;
        // hint L0/L2 for the first row of this window (global_prefetch_b8)
        {
            int hp = h0 < 0 ? 0 : h0;
            int wp = w0 < 0 ? 0 : w0;
            __builtin_prefetch(&lg[hp * WW + wp], 0, 3);
        }
        for (int i = 0; i < 4; ++i) {
            int h = h0 + i;
            if (h < 0 || h >= HH) continue;
            for (int j = 0; j < 4; ++j) {
                int w = w0 + j;
                if (w < 0 || w >= WW) continue;
                int lbl = lb[h * WW + w];
                bool m  = (lbl < NUM_CLASSES);
                float x = lg[h * WW + w];
                float t = (m && lbl == c) ? 1.f : 0.f;
                float s = 1.f / (1.f + expf(-x));
                prSum  += (m ? s : 0.f) + CLP_MIN;   // probs = sig*mask + clip
                labSum += t;                          // onehot*mask
                float bce = fmaxf(x, 0.f) - x * t + log1pf(expf(-fabsf(x)));
                bceLocal += m ? bce : 0.f;
                if (c == 0) vLocal += m ? 1u : 0u;    // count each pixel once
            }
        }
        labPool[idx] = labSum * (1.f / 16.f);  // count_include_pad=True
        prPool[idx]  = prSum  * (1.f / 16.f);
    }

    // block reduction -> global atomics
    __shared__ double sb[256];
    __shared__ unsigned int sv[256];
    sb[threadIdx.x] = (double)bceLocal;
    sv[threadIdx.x] = vLocal;
    __syncthreads();
    for (int off = 128; off > 0; off >>= 1) {
        if (threadIdx.x < off) {
            sb[threadIdx.x] += sb[threadIdx.x + off];
            sv[threadIdx.x] += sv[threadIdx.x + off];
        }
        __syncthreads();
    }
    if (threadIdx.x == 0) {
        atomicAdd(bceSum, sb[0]);
        atomicAdd(validCnt, sv[0]);
    }
}

// ---------------------------------------------------------------------------
// Kernel 2: per-(n,c) Gram matrices of the 9 shifted views via
// V_WMMA_F32_16X16X4_F32 (rows 9..15 zero-padded).  K domain padded to
// 127x128 so y=k>>7, x=k&127; phantom column x==127 masked by one multiply.
// KPAD = 16*1016 exactly -> no tail, no divergence in the hot loop.
// Centering folded in via row sums: cov = Gram - s s^T / K.
// 76 blocks x 16 waves; LDS-reduce, then f64 9x9 Schur + Cholesky logdet.
// ---------------------------------------------------------------------------
__global__ void __launch_bounds__(32 * GWAVES)
rmi_gram_kernel(const float* __restrict__ labPool,
                const float* __restrict__ prPool,
                double* __restrict__ logdetOut) {
    const int b  = blockIdx.x;                 // (n,c) pair
    const int PP = PHH * PWW;
    const float* la = labPool + (size_t)b * PP;
    const float* pr = prPool  + (size_t)b * PP;

    const int tid  = threadIdx.x;
    const int wave = tid >> 5;
    const int lane = tid & 31;
    const int d    = lane & 15;                    // matrix row (M)
    const bool rowOk = (d < HALF_D);
    const int dy   = rowOk ? d / 3 : 0;            // clamp pad rows: keep
    const int dx   = rowOk ? d % 3 : 0;            //   addresses in-bounds
    const int offd = dy * PWW + dx;
    const float rowScale = rowOk ? 1.f : 0.f;
    const int kOff = (lane >> 4) << 1;             // lanes>=16 hold K+2,K+3

    const int CHUNK = KPAD / GWAVES;               // 1016, multiple of 4
    const int NIT   = CHUNK / 4;                   // 254 iterations
    int k0 = wave * CHUNK + kOff;

    v8f accLL = {};  // la . la^T
    v8f accPP = {};  // pr . pr^T
    v8f accLP = {};  // la . pr^T
    float sLa = 0.f, sPr = 0.f;

    for (int it = 0; it < NIT; ++it) {
        int y = k0 >> 7;                           // padded-row index
        int x = k0 & 127;                          // even -> always <= 126
        int o = y * PWW + x + offd;
        float la0 = la[o];
        float la1 = la[o + 1];
        float pr0 = pr[o];
        float pr1 = pr[o + 1];
        float s1  = (x == 126) ? 0.f : rowScale;   // mask phantom column 127
        la0 *= rowScale;  la1 *= s1;
        pr0 *= rowScale;  pr1 *= s1;
        // A(M=lane%16) and B(N=lane%16) register images coincide for the
        // symmetric 16x16x4 layouts: one packed pair serves both operands.
        v2f aLa = {la0, la1};
        v2f aPr = {pr0, pr1};
        accLL = __builtin_amdgcn_wmma_f32_16x16x4_f32(false, aLa, false, aLa,
                                                      (short)0, accLL, false, false);
        accPP = __builtin_amdgcn_wmma_f32_16x16x4_f32(false, aPr, false, aPr,
                                                      (short)0, accPP, false, false);
        accLP = __builtin_amdgcn_wmma_f32_16x16x4_f32(false, aLa, false, aPr,
                                                      (short)0, accLP, false, false);
        sLa += la0 + la1;
        sPr += pr0 + pr1;
        k0 += 4;
    }

    // ---- LDS reduction across waves ----
    __shared__ float part[GWAVES][3][32][8];   // wave, matrix, lane, reg
    __shared__ float sums[GWAVES][2][32];
    __shared__ float gram[3][16][16];
    __shared__ double sRow[2][16];

    for (int r = 0; r < 8; ++r) {
        part[wave][0][lane][r] = accLL[r];
        part[wave][1][lane][r] = accPP[r];
        part[wave][2][lane][r] = accLP[r];
    }
    sums[wave][0][lane] = sLa;
    sums[wave][1][lane] = sPr;
    __syncthreads();

    if (tid < 256) {   // 256 slots = 32 lanes x 8 regs; C-layout -> (i,j)
        int lane2 = tid & 31;
        int r     = tid >> 5;
        int i = r + ((lane2 & 16) ? 8 : 0);
        int j = lane2 & 15;
        for (int m = 0; m < 3; ++m) {
            float v = 0.f;
            for (int w = 0; w < GWAVES; ++w) v += part[w][m][lane2][r];
            gram[m][i][j] = v;
        }
    }
    __syncthreads();
    if (tid < 16) {
        double a = 0.0, c2 = 0.0;
        for (int w = 0; w < GWAVES; ++w) {
            a  += (double)sums[w][0][tid] + (double)sums[w][0][tid + 16];
            c2 += (double)sums[w][1][tid] + (double)sums[w][1][tid + 16];
        }
        sRow[0][tid] = a;
        sRow[1][tid] = c2;
    }
    __syncthreads();

    // ---- serial f64 solve (9x9): Schur complement + Cholesky logdet ----
    if (tid == 0) {
        const double Kd = (double)KTOT;
        double laC[9][9], prC[9][9], lp[9][9];
        for (int i = 0; i < 9; ++i)
            for (int j = 0; j < 9; ++j) {
                laC[i][j] = (double)gram[0][i][j] - sRow[0][i] * sRow[0][j] / Kd;
                prC[i][j] = (double)gram[1][i][j] - sRow[1][i] * sRow[1][j] / Kd
                            + (i == j ? POS_ALPHA : 0.0);
                lp[i][j]  = (double)gram[2][i][j] - sRow[0][i] * sRow[1][j] / Kd;
            }
        // Cholesky of prC
        double L[9][9];
        for (int i = 0; i < 9; ++i) {
            for (int j = 0; j <= i; ++j) {
                double s = prC[i][j];
                for (int k = 0; k < j; ++k) s -= L[i][k] * L[j][k];
                L[i][j] = (i == j) ? sqrt(s) : s / L[j][j];
            }
        }
        // X = inv(prC) * lp^T  (column c0 solves prC x = lp[c0][:])
        double X[9][9];
        for (int c0 = 0; c0 < 9; ++c0) {
            double y[9], xv[9];
            for (int i = 0; i < 9; ++i) {
                double s = lp[c0][i];
                for (int j = 0; j < i; ++j) s -= L[i][j] * y[j];
                y[i] = s / L[i][i];
            }
            for (int i = 8; i >= 0; --i) {
                double s = y[i];
                for (int j = i + 1; j < 9; ++j) s -= L[j][i] * xv[j];
                xv[i] = s / L[i][i];
            }
            for (int i = 0; i < 9; ++i) X[i][c0] = xv[i];
        }
        // appro = laC - lp @ X + alpha I ; Cholesky logdet
        double A2[9][9];
        for (int i = 0; i < 9; ++i)
            for (int j = 0; j < 9; ++j) {
                double s = 0.0;
                for (int m = 0; m < 9; ++m) s += lp[i][m] * X[m][j];
                A2[i][j] = laC[i][j] - s + (i == j ? POS_ALPHA : 0.0);
            }
        double ldet = 0.0;
        double C2[9][9];
        for (int i = 0; i < 9; ++i) {
            for (int j = 0; j <= i; ++j) {
                double s = A2[i][j];
                for (int k = 0; k < j; ++k) s -= C2[i][k] * C2[j][k];
                if (i == j) {
                    double dg = sqrt(s);
                    C2[i][j] = dg;
                    ldet += log(dg);
                } else {
                    C2[i][j] = s / C2[j][j];
                }
            }
        }
        logdetOut[b] = 2.0 * ldet;
    }
}

// ---------------------------------------------------------------------------
// Kernel 3: scalar finalize
// ---------------------------------------------------------------------------
__global__ void rmi_finalize_kernel(const double* __restrict__ bceSum,
                                    const unsigned int* __restrict__ validCnt,
                                    const double* __restrict__ logdet,
                                    float* __restrict__ out) {
    if (threadIdx.x == 0 && blockIdx.x == 0) {
        double bce_loss = *bceSum / ((double)(*validCnt) + 1.0);
        float rmi = 0.f;
        for (int c = 0; c < NUM_CLASSES; ++c) {
            double m = 0.0;
            for (int n = 0; n < NN; ++n) m += 0.5 * logdet[n * NUM_CLASSES + c];
            m /= (double)NN;
            rmi += (float)m / (float)HALF_D;   // astype(f32) then / HALF_D
        }
        out[0] = (float)(0.5 * bce_loss + 0.5 * (double)rmi);
    }
}

// ---------------------------------------------------------------------------
extern "C" void kernel_launch(void* const* d_in, const int* in_sizes, int n_in,
                              void* d_out, int out_size, void* d_ws, size_t ws_size,
                              hipStream_t stream) {
    (void)in_sizes; (void)n_in; (void)out_size; (void)ws_size;
    const float* logits = (const float*)d_in[0];
    const int*   labels = (const int*)d_in[1];

    char* ws = (char*)d_ws;
    double*       bceSum   = (double*)(ws + 0);
    unsigned int* validCnt = (unsigned int*)(ws + 8);
    double*       logdet   = (double*)(ws + 16);           // 76 doubles

    const size_t PP = (size_t)PHH * PWW;
    const size_t P  = (size_t)NPAIR * PP;                  // 1,264,716 floats
    size_t poolBytes = (P * sizeof(float) + 255) & ~(size_t)255; // +padding
    float* labPool = (float*)(ws + 1024);
    float* prPool  = (float*)(ws + 1024 + poolBytes);

    rmi_init_kernel<<<1, 1, 0, stream>>>(bceSum, validCnt);

    int tot = (int)P;
    rmi_bce_pool_kernel<<<(tot + 255) / 256, 256, 0, stream>>>(
        logits, labels, labPool, prPool, bceSum, validCnt);

    rmi_gram_kernel<<<NPAIR, 32 * GWAVES, 0, stream>>>(labPool, prPool, logdet);

    rmi_finalize_kernel<<<1, 32, 0, stream>>>(bceSum, validCnt, logdet,
                                              (float*)d_out);
}